// GCN_17008070492327
// MI455X (gfx1250) — compile-verified
//
#include <hip/hip_runtime.h>
#include <math.h>

typedef float v2f __attribute__((ext_vector_type(2)));
typedef float v8f __attribute__((ext_vector_type(8)));

#define TPB 256

__device__ __forceinline__ void atomAddF32(float* p, float v) {
    unsafeAtomicAdd(p, v);   // lowers to global_atomic_add_f32 (no CAS loop)
}

// ---------------------------------------------------------------- deg init
__global__ void k_init_deg(float* __restrict__ deg, int n) {
    int i = blockIdx.x * blockDim.x + threadIdx.x;
    if (i < n) deg[i] = 1.0f;                 // self-loop contributes 1
}

// ---------------------------- degree accumulation + int64 -> int32 convert
__global__ void k_deg_convert(const long long* __restrict__ ei, int E,
                              int* __restrict__ src32, int* __restrict__ dst32,
                              float* __restrict__ deg) {
    int e = blockIdx.x * blockDim.x + threadIdx.x;
    if (e >= E) return;
    int s = (int)ei[e];
    int d = (int)ei[(long long)E + e];
    src32[e] = s;
    dst32[e] = d;
    atomAddF32(&deg[d], 1.0f);
}

// ---------------------------------------------------------------- deg -> dinv
__global__ void k_dinv(float* __restrict__ deg, int n) {
    int i = blockIdx.x * blockDim.x + threadIdx.x;
    if (i >= n) return;
    float dg = deg[i];
    deg[i] = (dg > 0.0f) ? rsqrtf(dg) : 0.0f;
}

// ---------------- layer-1 GEMM: H1 = x(Nx3) @ W1(3x16) via WMMA f32 16x16x4
// Full tiles only (tail handled by k_gemm1_tail). Also writes the self-loop
// term AGG1 = H1 * dinv^2 so no separate zero-init pass is needed.
__global__ void k_gemm1(const float* __restrict__ x, const float* __restrict__ W1,
                        const float* __restrict__ dinv,
                        float* __restrict__ H1, float* __restrict__ AGG1,
                        int n_tiles) {
    int wave = blockIdx.x * (blockDim.x >> 5) + (threadIdx.x >> 5);
    if (wave >= n_tiles) return;              // uniform across the wave
    int lane = threadIdx.x & 31;
    int half = lane >> 4;                     // 0: lanes 0-15, 1: lanes 16-31
    int lcol = lane & 15;                     // M for A, N for B/C/D
    int n0   = wave * 16;

    // A: 16x4 f32. half0 holds K={0,1}, half1 holds K={2,3(pad=0)}
    const float* xr = x + (n0 + lcol) * 3;
    v2f a;
    a.x = half ? xr[2] : xr[0];
    a.y = half ? 0.0f  : xr[1];

    // B: 4x16 f32; W1 is [3,16] row-major, K=3 row is zero padding
    v2f b;
    b.x = half ? W1[2 * 16 + lcol] : W1[0 * 16 + lcol];
    b.y = half ? 0.0f              : W1[1 * 16 + lcol];

    v8f c = {};
    c = __builtin_amdgcn_wmma_f32_16x16x4_f32(false, a, false, b,
                                              (short)0, c, false, false);

    // dinv for this lane's 8 rows (contiguous): rows n0+8*half .. +7
    const float4* dp = (const float4*)(dinv + n0 + 8 * half);
    float4 dv0 = dp[0], dv1 = dp[1];
    float dd[8] = {dv0.x, dv0.y, dv0.z, dv0.w, dv1.x, dv1.y, dv1.z, dv1.w};

    // D layout: VGPR r -> row n0 + 8*half + r, col lcol (coalesced b32 rows)
    int base = (n0 + 8 * half) * 16 + lcol;
#pragma unroll
    for (int r = 0; r < 8; ++r) {
        float v = c[r];
        H1[base + r * 16]   = v;
        AGG1[base + r * 16] = v * dd[r] * dd[r];
    }
}

// scalar tail for rows not covered by full 16-row tiles (0 rows for N=100000)
__global__ void k_gemm1_tail(const float* __restrict__ x, const float* __restrict__ W1,
                             const float* __restrict__ dinv,
                             float* __restrict__ H1, float* __restrict__ AGG1,
                             int row0, int N) {
    int t = row0 + (blockIdx.x * blockDim.x + threadIdx.x) / 16;
    int f = threadIdx.x & 15;
    if (t >= N) return;
    float v = x[t*3+0]*W1[f] + x[t*3+1]*W1[16+f] + x[t*3+2]*W1[32+f];
    float di = dinv[t];
    H1[t * 16 + f]   = v;
    AGG1[t * 16 + f] = v * di * di;
}

// --------------- layer-1 edge scatter: AGG1[dst] += H1[src] * dinv_s*dinv_d
__global__ void k_scatter1(const int* __restrict__ src32, const int* __restrict__ dst32,
                           const float* __restrict__ dinv,
                           const float* __restrict__ H1, float* __restrict__ AGG1,
                           int E) {
    int t = blockIdx.x * blockDim.x + threadIdx.x;
    int e = t >> 2;
    if (e >= E) return;
    int q = (t & 3) * 4;                      // 4 lanes per edge, 4 feats each
    int s = src32[e], d = dst32[e];
    float w = dinv[s] * dinv[d];
    float4 v = *(const float4*)(H1 + s * 16 + q);
    float* ap = AGG1 + d * 16 + q;
    atomAddF32(ap + 0, v.x * w);
    atomAddF32(ap + 1, v.y * w);
    atomAddF32(ap + 2, v.z * w);
    atomAddF32(ap + 3, v.w * w);
}

// ------------------------------------------- bias + ReLU, in place on AGG1
__global__ void k_relu_bias(float* __restrict__ AGG1, const float* __restrict__ b1,
                            int total) {
    int i = blockIdx.x * blockDim.x + threadIdx.x;
    if (i >= total) return;
    float v = AGG1[i] + b1[i & 15];
    AGG1[i] = v > 0.0f ? v : 0.0f;
}

// -------- layer-2 GEMM: H2 = X1(Nx16) @ W2(16x7 padded to 16) via 4x WMMA
__global__ void k_gemm2(const float* __restrict__ X1, const float* __restrict__ W2,
                        const float* __restrict__ dinv,
                        float* __restrict__ H2, float* __restrict__ AGG2,
                        int n_tiles) {
    int wave = blockIdx.x * (blockDim.x >> 5) + (threadIdx.x >> 5);
    if (wave >= n_tiles) return;              // uniform across the wave
    int lane = threadIdx.x & 31;
    int half = lane >> 4;
    int lcol = lane & 15;
    int n0   = wave * 16;

    const float* ar = X1 + (n0 + lcol) * 16 + 2 * half;
    bool valid = (lcol < 7);

    v8f c = {};
#pragma unroll
    for (int kb = 0; kb < 4; ++kb) {
        int k0 = 4 * kb + 2 * half;           // this half's K pair
        v2f a = *(const v2f*)(ar + 4 * kb);   // 8B-aligned global_load_b64
        v2f b;                                // W2 is [16,7] row-major
        b.x = valid ? W2[(k0    ) * 7 + lcol] : 0.0f;
        b.y = valid ? W2[(k0 + 1) * 7 + lcol] : 0.0f;
        c = __builtin_amdgcn_wmma_f32_16x16x4_f32(false, a, false, b,
                                                  (short)0, c, false, false);
    }

    if (lcol < 8) {                           // single predicate, 8-col rows
        const float4* dp = (const float4*)(dinv + n0 + 8 * half);
        float4 dv0 = dp[0], dv1 = dp[1];
        float dd[8] = {dv0.x, dv0.y, dv0.z, dv0.w, dv1.x, dv1.y, dv1.z, dv1.w};
        int base = (n0 + 8 * half) * 8 + lcol;
#pragma unroll
        for (int r = 0; r < 8; ++r) {
            float v = valid ? c[r] : 0.0f;    // col 7 stays zero
            H2[base + r * 8]   = v;
            AGG2[base + r * 8] = v * dd[r] * dd[r];
        }
    }
}

__global__ void k_gemm2_tail(const float* __restrict__ X1, const float* __restrict__ W2,
                             const float* __restrict__ dinv,
                             float* __restrict__ H2, float* __restrict__ AGG2,
                             int row0, int N) {
    int t = row0 + (blockIdx.x * blockDim.x + threadIdx.x) / 8;
    int f = threadIdx.x & 7;
    if (t >= N) return;
    float v = 0.0f;
    if (f < 7) {
        for (int k = 0; k < 16; ++k) v += X1[t * 16 + k] * W2[k * 7 + f];
    }
    float di = dinv[t];
    H2[t * 8 + f]   = v;
    AGG2[t * 8 + f] = v * di * di;
}

// ----------------------- layer-2 edge scatter (8-float rows, 2 lanes/edge)
__global__ void k_scatter2(const int* __restrict__ src32, const int* __restrict__ dst32,
                           const float* __restrict__ dinv,
                           const float* __restrict__ H2, float* __restrict__ AGG2,
                           int E) {
    int t = blockIdx.x * blockDim.x + threadIdx.x;
    int e = t >> 1;
    if (e >= E) return;
    int q = (t & 1) * 4;
    int s = src32[e], d = dst32[e];
    float w = dinv[s] * dinv[d];
    float4 v = *(const float4*)(H2 + s * 8 + q);
    float* ap = AGG2 + d * 8 + q;
    atomAddF32(ap + 0, v.x * w);
    atomAddF32(ap + 1, v.y * w);
    atomAddF32(ap + 2, v.z * w);
    atomAddF32(ap + 3, v.w * w);              // pad col adds 0, harmless
}

// --------------------------------------------- bias + log_softmax (7 cls)
__global__ void k_logsm(const float* __restrict__ AGG2, const float* __restrict__ b2,
                        float* __restrict__ out, int N) {
    int i = blockIdx.x * blockDim.x + threadIdx.x;
    if (i >= N) return;
    float4 p0 = *(const float4*)(AGG2 + i * 8);
    float4 p1 = *(const float4*)(AGG2 + i * 8 + 4);
    float v[7] = {p0.x, p0.y, p0.z, p0.w, p1.x, p1.y, p1.z};
    float m = -3.402823466e38f;
#pragma unroll
    for (int cidx = 0; cidx < 7; ++cidx) {
        v[cidx] += b2[cidx];
        m = fmaxf(m, v[cidx]);
    }
    float s = 0.0f;
#pragma unroll
    for (int cidx = 0; cidx < 7; ++cidx) s += __expf(v[cidx] - m);
    float lse = m + __logf(s);
#pragma unroll
    for (int cidx = 0; cidx < 7; ++cidx) out[i * 7 + cidx] = v[cidx] - lse;
}

static inline size_t align256(size_t v) { return (v + 255) & ~(size_t)255; }

extern "C" void kernel_launch(void* const* d_in, const int* in_sizes, int n_in,
                              void* d_out, int out_size, void* d_ws, size_t ws_size,
                              hipStream_t stream) {
    const float*     x   = (const float*)d_in[0];
    const long long* ei  = (const long long*)d_in[1];
    const float*     W1  = (const float*)d_in[2];
    const float*     b1  = (const float*)d_in[3];
    const float*     W2  = (const float*)d_in[4];
    const float*     b2  = (const float*)d_in[5];
    float*           out = (float*)d_out;

    const int N = in_sizes[0] / 3;
    const int E = in_sizes[1] / 2;

    // workspace carve-up
    char* ws = (char*)d_ws;
    size_t off = 0;
    float* dinv  = (float*)(ws + off); off += align256((size_t)N * 4);
    int*   src32 = (int*)  (ws + off); off += align256((size_t)E * 4);
    int*   dst32 = (int*)  (ws + off); off += align256((size_t)E * 4);
    float* H1    = (float*)(ws + off); off += align256((size_t)N * 16 * 4);
    float* AGG1  = (float*)(ws + off); off += align256((size_t)N * 16 * 4);
    float* H2    = (float*)(ws + off); off += align256((size_t)N * 8 * 4);
    float* AGG2  = (float*)(ws + off); off += align256((size_t)N * 8 * 4);
    (void)ws_size;

    const int n_tiles = N / 16;               // full tiles (6250, exact here)
    const int rem_row0 = n_tiles * 16;
    const int rem = N - rem_row0;
    const int waves_per_blk = TPB / 32;
    const int gemm_blocks = (n_tiles + waves_per_blk - 1) / waves_per_blk;

    k_init_deg   <<<(N + TPB - 1) / TPB, TPB, 0, stream>>>(dinv, N);
    k_deg_convert<<<(E + TPB - 1) / TPB, TPB, 0, stream>>>(ei, E, src32, dst32, dinv);
    k_dinv       <<<(N + TPB - 1) / TPB, TPB, 0, stream>>>(dinv, N);

    k_gemm1      <<<gemm_blocks, TPB, 0, stream>>>(x, W1, dinv, H1, AGG1, n_tiles);
    if (rem > 0)
        k_gemm1_tail<<<(rem * 16 + TPB - 1) / TPB, TPB, 0, stream>>>(x, W1, dinv,
                                                                     H1, AGG1, rem_row0, N);
    {
        long long tt = (long long)E * 4;
        k_scatter1<<<(int)((tt + TPB - 1) / TPB), TPB, 0, stream>>>(src32, dst32, dinv,
                                                                    H1, AGG1, E);
    }
    k_relu_bias  <<<(N * 16 + TPB - 1) / TPB, TPB, 0, stream>>>(AGG1, b1, N * 16);

    k_gemm2      <<<gemm_blocks, TPB, 0, stream>>>(AGG1, W2, dinv, H2, AGG2, n_tiles);
    if (rem > 0)
        k_gemm2_tail<<<(rem * 8 + TPB - 1) / TPB, TPB, 0, stream>>>(AGG1, W2, dinv,
                                                                    H2, AGG2, rem_row0, N);
    {
        long long tt = (long long)E * 2;
        k_scatter2<<<(int)((tt + TPB - 1) / TPB), TPB, 0, stream>>>(src32, dst32, dinv,
                                                                    H2, AGG2, E);
    }
    k_logsm      <<<(N + TPB - 1) / TPB, TPB, 0, stream>>>(AGG2, b2, out, N);
}